// DecoderAttention_29051158790749
// MI455X (gfx1250) — compile-verified
//
#include <hip/hip_runtime.h>

// Problem dims (compile-time)
constexpr int Bb = 4;
constexpr int Ss = 2048;
constexpr int Hh = 1024;
constexpr int NHh = 16;
constexpr int HDd = 64;            // head dim
constexpr float kScale = 0.125f;   // 64^-0.5

typedef float v2f __attribute__((ext_vector_type(2)));
typedef float v8f __attribute__((ext_vector_type(8)));

static __device__ __forceinline__ v8f wmma4(v2f a, v2f b, v8f c) {
  return __builtin_amdgcn_wmma_f32_16x16x4_f32(false, a, false, b,
                                               (short)0, c, false, false);
}

// ---------------------------------------------------------------------------
// GEMM: Y[M,N] = X[M,K] @ W[N,K]^T + bias[N]   (torch Linear)
// One wave computes a 32x32 macro-tile = 2x2 WMMA tiles: 2 A-frags + 2 B-frags
// feed 4 v_wmma_f32_16x16x4_f32 per k-step (8 FLOP/byte from L2).
// A-frag layout (16x4 f32): lane m = l&15; VGPR0/1 hold K = 2*(l>>4), +1.
// C/D layout: c[i] -> row (i + 8*(l>>4)), col (l&15).
// ---------------------------------------------------------------------------
__global__ __launch_bounds__(256) void gemm_xwt_bias(
    const float* __restrict__ X, const float* __restrict__ W,
    const float* __restrict__ bias, float* __restrict__ Y,
    int M, int N, int K)
{
  const int lane = threadIdx.x & 31;
  const int wave = threadIdx.x >> 5;
  const int tN = N >> 5;                       // 32-wide macro tiles
  const int tile = blockIdx.x * 8 + wave;
  const int totalTiles = (M >> 5) * tN;
  if (tile >= totalTiles) return;              // wave-uniform: EXEC stays all-1s
  const int mt = tile / tN, nt = tile % tN;
  const int m0 = mt << 5, n0 = nt << 5;
  const int lo = lane & 15, hi = lane >> 4;

  const float* x0 = X + (size_t)(m0 + lo) * K;        // A rows m0..m0+15
  const float* x1 = X + (size_t)(m0 + 16 + lo) * K;   // A rows m0+16..m0+31
  const float* w0 = W + (size_t)(n0 + lo) * K;        // B = W^T cols n0..+15
  const float* w1 = W + (size_t)(n0 + 16 + lo) * K;   // B cols n0+16..+31

  v8f c00 = {}, c01 = {}, c10 = {}, c11 = {};
  for (int k0 = 0; k0 < K; k0 += 4) {
    if ((k0 & 63) == 0 && k0 + 64 < K) {              // near-scope prefetch
      __builtin_prefetch(x0 + k0 + 64, 0, 3);
      __builtin_prefetch(x1 + k0 + 64, 0, 3);
      __builtin_prefetch(w0 + k0 + 64, 0, 3);
      __builtin_prefetch(w1 + k0 + 64, 0, 3);
    }
    const int ko = k0 + 2 * hi;
    v2f a0 = *(const v2f*)(x0 + ko);
    v2f a1 = *(const v2f*)(x1 + ko);
    v2f b0 = *(const v2f*)(w0 + ko);
    v2f b1 = *(const v2f*)(w1 + ko);
    c00 = wmma4(a0, b0, c00);
    c01 = wmma4(a0, b1, c01);
    c10 = wmma4(a1, b0, c10);
    c11 = wmma4(a1, b1, c11);
  }
  const float bv0 = bias[n0 + lo];
  const float bv1 = bias[n0 + 16 + lo];
  float* y0 = Y + (size_t)(m0 + 8 * hi) * N + n0 + lo;
  float* y1 = Y + (size_t)(m0 + 16 + 8 * hi) * N + n0 + lo;
  #pragma unroll
  for (int i = 0; i < 8; ++i) {
    y0[(size_t)i * N]      = c00[i] + bv0;
    y0[(size_t)i * N + 16] = c01[i] + bv1;
    y1[(size_t)i * N]      = c10[i] + bv0;
    y1[(size_t)i * N + 16] = c11[i] + bv1;
  }
}

// ---------------------------------------------------------------------------
// scores = scale * Q K^T with causal mask, written pre-softmax into attn.
// Write-bound stage (1.07 GB): tiles above the diagonal are filled without
// compute; Q/K fragment reads are L2-resident.
// ---------------------------------------------------------------------------
__global__ __launch_bounds__(256) void attn_scores(
    const float* __restrict__ Q, const float* __restrict__ Kmat,
    float* __restrict__ attn)
{
  const int lane = threadIdx.x & 31;
  const int wave = threadIdx.x >> 5;
  const int QT = Ss / 16, KT = Ss / 16;
  const long tile = (long)blockIdx.x * 8 + wave;   // (bh, qt, kt)
  const int kt = (int)(tile % KT);
  const long t2 = tile / KT;
  const int qt = (int)(t2 % QT);
  const int bh = (int)(t2 / QT);
  const int b = bh / NHh, h = bh % NHh;
  const int q0 = qt << 4, k0 = kt << 4;
  const int lo = lane & 15, hi = lane >> 4;

  float* atile = attn + ((size_t)bh * Ss + q0) * Ss + k0;

  if (kt > qt) {                                   // fully masked tile
    #pragma unroll
    for (int i = 0; i < 8; ++i)
      atile[(size_t)(i + 8 * hi) * Ss + lo] = -10000.0f;
    return;
  }

  const float* qrow = Q    + (size_t)(b * Ss + q0 + lo) * Hh + h * HDd;
  const float* krow = Kmat + (size_t)(b * Ss + k0 + lo) * Hh + h * HDd;

  v8f c = {};
  #pragma unroll
  for (int d0 = 0; d0 < HDd; d0 += 4) {
    v2f a  = *(const v2f*)(qrow + d0 + 2 * hi);
    v2f bb = *(const v2f*)(krow + d0 + 2 * hi);   // B[d][n] = K[k0+n][d]
    c = wmma4(a, bb, c);
  }
  #pragma unroll
  for (int i = 0; i < 8; ++i) {
    const int r  = q0 + i + 8 * hi;
    const int cc = k0 + lo;
    atile[(size_t)(i + 8 * hi) * Ss + lo] =
        (cc > r) ? -10000.0f : c[i] * kScale;
  }
}

// ---------------------------------------------------------------------------
// In-place row softmax over the last axis (length S). One wave per row,
// wave32 __shfl_xor butterfly reductions.
// ---------------------------------------------------------------------------
__global__ __launch_bounds__(256) void softmax_rows(float* __restrict__ attn)
{
  const int lane = threadIdx.x & 31;
  const int wave = threadIdx.x >> 5;
  const size_t row = (size_t)blockIdx.x * 8 + wave;   // < B*NH*S
  float* p = attn + row * Ss;

  float m = -3.4e38f;
  for (int j = lane; j < Ss; j += 32) m = fmaxf(m, p[j]);
  #pragma unroll
  for (int off = 16; off > 0; off >>= 1) m = fmaxf(m, __shfl_xor(m, off, 32));

  float sum = 0.f;
  for (int j = lane; j < Ss; j += 32) {
    float e = __expf(p[j] - m);
    p[j] = e;
    sum += e;
  }
  #pragma unroll
  for (int off = 16; off > 0; off >>= 1) sum += __shfl_xor(sum, off, 32);

  const float inv = 1.0f / sum;
  for (int j = lane; j < Ss; j += 32) p[j] *= inv;
}

// ---------------------------------------------------------------------------
// out_heads = attn @ V. One wave produces the full 16x64 strip for a q-tile:
// the attn A-fragment (the dominant, 1.07 GB stream) is loaded ONCE and
// applied against 4 V B-fragments (4 WMMAs per k-step). K-loop truncated at
// the causal boundary (attn rows are zero past q0+16).
// ---------------------------------------------------------------------------
__global__ __launch_bounds__(256) void attn_v(
    const float* __restrict__ attn, const float* __restrict__ Vmat,
    float* __restrict__ OH)
{
  const int lane = threadIdx.x & 31;
  const int wave = threadIdx.x >> 5;
  const int QT = Ss / 16;
  const int tile = blockIdx.x * 8 + wave;           // (bh, qt)
  const int qt = tile % QT;
  const int bh = tile / QT;
  const int b = bh / NHh, h = bh % NHh;
  const int q0 = qt << 4;
  const int lo = lane & 15, hi = lane >> 4;

  const float* arow  = attn + ((size_t)bh * Ss + q0 + lo) * Ss;
  const float* vbase = Vmat + (size_t)b * Ss * Hh + h * HDd + lo;  // col n=lo (+16j)

  v8f c0 = {}, c1 = {}, c2 = {}, c3 = {};
  const int kend = q0 + 16;                         // causal: rest is zero
  for (int k0 = 0; k0 < kend; k0 += 4) {
    v2f a = *(const v2f*)(arow + k0 + 2 * hi);
    const float* vk0 = vbase + (size_t)(k0 + 2 * hi) * Hh;      // B row k
    const float* vk1 = vbase + (size_t)(k0 + 2 * hi + 1) * Hh;  // B row k+1
    v2f b0 = {vk0[0],  vk1[0]};
    v2f b1 = {vk0[16], vk1[16]};
    v2f b2 = {vk0[32], vk1[32]};
    v2f b3 = {vk0[48], vk1[48]};
    c0 = wmma4(a, b0, c0);
    c1 = wmma4(a, b1, c1);
    c2 = wmma4(a, b2, c2);
    c3 = wmma4(a, b3, c3);
  }
  float* out = OH + (size_t)(b * Ss + q0 + 8 * hi) * Hh + h * HDd + lo;
  #pragma unroll
  for (int i = 0; i < 8; ++i) {
    out[(size_t)i * Hh]      = c0[i];
    out[(size_t)i * Hh + 16] = c1[i];
    out[(size_t)i * Hh + 32] = c2[i];
    out[(size_t)i * Hh + 48] = c3[i];
  }
}

// ---------------------------------------------------------------------------
extern "C" void kernel_launch(void* const* d_in, const int* in_sizes, int n_in,
                              void* d_out, int out_size, void* d_ws, size_t ws_size,
                              hipStream_t stream) {
  (void)in_sizes; (void)n_in; (void)out_size; (void)ws_size;
  const float* x  = (const float*)d_in[0];
  // d_in[1] is the causal mask (int32) — handled analytically, never read.
  const float* Wq = (const float*)d_in[2];
  const float* bq = (const float*)d_in[3];
  const float* Wk = (const float*)d_in[4];
  const float* bk = (const float*)d_in[5];
  const float* Wv = (const float*)d_in[6];
  const float* bv = (const float*)d_in[7];
  const float* Wo = (const float*)d_in[8];
  const float* bo = (const float*)d_in[9];

  float* out  = (float*)d_out;                       // [B,S,H]
  float* attn = out + (size_t)Bb * Ss * Hh;          // [B,NH,S,S]

  float* q  = (float*)d_ws;
  float* k  = q  + (size_t)Bb * Ss * Hh;
  float* v  = k  + (size_t)Bb * Ss * Hh;
  float* oh = v  + (size_t)Bb * Ss * Hh;

  const int M = Bb * Ss;                                           // 8192
  const int gemmTiles  = (M / 32) * (Hh / 32);                     // 8192
  const int gemmBlocks = gemmTiles / 8;                            // 1024
  const int scoreBlocks = (Bb * NHh * (Ss / 16) * (Ss / 16)) / 8;  // 131072
  const int smaxBlocks  = (Bb * NHh * Ss) / 8;                     // 16384
  const int avBlocks    = (Bb * NHh * (Ss / 16)) / 8;              // 1024

  gemm_xwt_bias<<<gemmBlocks, 256, 0, stream>>>(x, Wq, bq, q, M, Hh, Hh);
  gemm_xwt_bias<<<gemmBlocks, 256, 0, stream>>>(x, Wk, bk, k, M, Hh, Hh);
  gemm_xwt_bias<<<gemmBlocks, 256, 0, stream>>>(x, Wv, bv, v, M, Hh, Hh);
  attn_scores  <<<scoreBlocks, 256, 0, stream>>>(q, k, attn);
  softmax_rows <<<smaxBlocks, 256, 0, stream>>>(attn);
  attn_v       <<<avBlocks, 256, 0, stream>>>(attn, v, oh);
  gemm_xwt_bias<<<gemmBlocks, 256, 0, stream>>>(oh, Wo, bo, out, M, Hh, Hh);
}